// HighFreqSuppressionLoss_49117245997517
// MI455X (gfx1250) — compile-verified
//
#include <hip/hip_runtime.h>
#include <hip/hip_bf16.h>

typedef __attribute__((ext_vector_type(2))) float v2f;
typedef __attribute__((ext_vector_type(8))) float v8f;
typedef __attribute__((ext_vector_type(4))) unsigned int u32x4;
typedef __attribute__((ext_vector_type(8))) int i32x8;
typedef __attribute__((ext_vector_type(4))) int i32x4;

#define IMG_N   512
#define IMG_PIX (512*512)
#define CHUNK   16
#define HF_CNT  245760.0f   // 512*512 - 128*128

#define A_PITCH   17        // 16-float rows + 1 pad  (conflict-free A frags)
#define BP_ROW    136       // 128-float rows + 8 pad (lane-half bank offset 16)
#define BP_COL    72        // 64-float rows + 8 pad  (lane-half bank offset 16)

#if __has_builtin(__builtin_amdgcn_tensor_load_to_lds)
#  define HF_HAS_TDM 1
#  if __has_include(<hip/amd_detail/amd_gfx1250_TDM.h>)
#    define HF_TDM6 1       // amdgpu-toolchain: 6-arg builtin
#  endif
#else
#  define HF_HAS_TDM 0
#endif

static __device__ __forceinline__ v8f hf_zero8() {
    v8f z = {0.f,0.f,0.f,0.f,0.f,0.f,0.f,0.f};
    return z;
}

// ---------------------------------------------------------------------------
// TDM: DMA a 2-D tile (tile_w x tile_h elems, f32) from a row-major matrix
// into LDS, with hardware row padding to the requested LDS pitch.
// pad codes: interval 0=2,1=4,2=8,3=16,4=32,5=64,6=128,7=256 dwords
//            amount   0=1 ... 7=8 dwords
// ---------------------------------------------------------------------------
static __device__ __forceinline__ void hf_tdm_load2d(
    const float* gaddr, unsigned lds_off,
    unsigned tile_w, unsigned tile_h,
    unsigned tensor_w, unsigned tensor_h, unsigned row_stride,
    unsigned pad_int_code, unsigned pad_amt_code)
{
#if HF_HAS_TDM
    unsigned long long ga = (unsigned long long)gaddr;
    u32x4 g0;
    g0.x = 1u;                                            // count=1 (valid D#)
    g0.y = lds_off;                                       // lds_addr (bytes)
    g0.z = (unsigned)(ga & 0xFFFFFFFFu);                  // global_addr[31:0]
    g0.w = (unsigned)((ga >> 32) & 0x01FFFFFFu)           // global_addr[56:32]
         | 0x80000000u;                                   // type = 2 ("image")
    i32x8 g1;
    g1[0] = (int)((2u << 16)                              // data_size = 4B
                | (1u << 20)                              // pad_enable
                | (pad_int_code << 22)
                | (pad_amt_code << 25));
    g1[1] = (int)(tensor_w << 16);                        // tensor_dim0[15:0]
    g1[2] = (int)((tensor_w >> 16) | (tensor_h << 16));   // dim0 hi | dim1 lo
    g1[3] = (int)((tensor_h >> 16) | (tile_w  << 16));    // dim1 hi | tile_dim0
    g1[4] = (int)tile_h;                                  // tile_dim1 (tile_dim2=0)
    g1[5] = (int)row_stride;                              // tensor_dim0_stride lo
    g1[6] = 0;
    g1[7] = 0;
    i32x4 gz = {0, 0, 0, 0};
#  if defined(HF_TDM6)
    i32x8 gz8 = {0,0,0,0,0,0,0,0};
    __builtin_amdgcn_tensor_load_to_lds(g0, g1, gz, gz, gz8, 0);
#  else
    __builtin_amdgcn_tensor_load_to_lds(g0, g1, gz, gz, 0);
#  endif
#else
    (void)gaddr; (void)lds_off; (void)tile_w; (void)tile_h;
    (void)tensor_w; (void)tensor_h; (void)row_stride;
    (void)pad_int_code; (void)pad_amt_code;
#endif
}

static __device__ __forceinline__ unsigned hf_lds_off(const void* p) {
    return (unsigned)(unsigned long long)p;   // low 32 bits = LDS byte offset
}

// ---------------------------------------------------------------------------
// DFT matrix: W[j,k] = exp(-2*pi*i * j*k / 512)
// ---------------------------------------------------------------------------
__global__ __launch_bounds__(256) void hf_initW(float* __restrict__ Wr,
                                                float* __restrict__ Wi) {
    int idx = blockIdx.x * 256 + threadIdx.x;      // 0 .. 262143
    int j = idx >> 9;
    int k = idx & 511;
    int p = (j * k) & 511;                          // phase mod 512
    float ang = -6.28318530717958647692f * (float)p * (1.0f / 512.0f);
    Wr[idx] = cosf(ang);
    Wi[idx] = sinf(ang);
}

// ---------------------------------------------------------------------------
// RGB [-1,1] -> grayscale [0,1] for a 16-image chunk
// ---------------------------------------------------------------------------
__global__ __launch_bounds__(256) void hf_gray(const float* __restrict__ src,
                                               int imgBase,
                                               float* __restrict__ G) {
    size_t tid = (size_t)blockIdx.x * 256 + threadIdx.x;
    int imgLocal = (int)(tid >> 18);
    int p = (int)(tid & (IMG_PIX - 1));
    size_t base = ((size_t)(imgBase + imgLocal)) * 3u * IMG_PIX;
    float r = src[base + p];
    float g = src[base + IMG_PIX + p];
    float b = src[base + 2 * IMG_PIX + p];
    G[tid] = (r * 0.299f + g * 0.587f + b * 0.114f + 1.0f) * 0.5f;
}

// ---------------------------------------------------------------------------
// Row pass: Y = G * W   (512^3 real GEMM; z-parity selects Wr / Wi)
// Block tile 128x128, BK=16, 8 waves (2x4), wave tile 64x32 -> 8 wmma accs
// Tiles arrive via TDM (double-buffered LDS, TENSORcnt-pipelined)
// ---------------------------------------------------------------------------
__global__ __launch_bounds__(256) void hf_rowpass(const float* __restrict__ G,
                                                  const float* __restrict__ Wr,
                                                  const float* __restrict__ Wi,
                                                  float* __restrict__ Yr,
                                                  float* __restrict__ Yi) {
    __shared__ float As[2][128 * A_PITCH];
    __shared__ float Bs[2][16 * BP_ROW];

    const int tid  = threadIdx.x;
    const int z    = blockIdx.z;
    const int img  = z >> 1;
    const float* A  = G + (size_t)img * IMG_PIX;
    const float* Bm = (z & 1) ? Wi : Wr;
    float* Y = ((z & 1) ? Yi : Yr) + (size_t)img * IMG_PIX;

    const int m0 = blockIdx.y * 128;
    const int n0 = blockIdx.x * 128;

    const int lane  = tid & 31;
    const int wave  = tid >> 5;
    const int waveM = wave >> 2;   // 0..1
    const int waveN = wave & 3;    // 0..3
    const int half  = lane >> 4;   // 0/1
    const int r     = lane & 15;

    v8f acc[4][2];
#pragma unroll
    for (int i = 0; i < 4; i++)
#pragma unroll
        for (int j = 0; j < 2; j++) acc[i][j] = hf_zero8();

#if HF_HAS_TDM
    const bool issuer = (wave == 0);
    if (issuer) {   // prime buffer 0
        hf_tdm_load2d(A  + (size_t)m0 * 512,      hf_lds_off(&As[0][0]),
                      16, 128, 512, 512, 512, 3, 0);
        hf_tdm_load2d(Bm + n0,                    hf_lds_off(&Bs[0][0]),
                      128, 16, 512, 512, 512, 6, 7);
    }
#endif

    for (int k0 = 0; k0 < 512; k0 += 16) {
        const int buf = (k0 >> 4) & 1;
#if HF_HAS_TDM
        if (issuer) {
            if (k0 + 16 < 512) {
                hf_tdm_load2d(A  + (size_t)m0 * 512 + (k0 + 16),
                              hf_lds_off(&As[buf ^ 1][0]),
                              16, 128, 512, 512, 512, 3, 0);
                hf_tdm_load2d(Bm + (size_t)(k0 + 16) * 512 + n0,
                              hf_lds_off(&Bs[buf ^ 1][0]),
                              128, 16, 512, 512, 512, 6, 7);
                __builtin_amdgcn_s_wait_tensorcnt(2);   // next pair in flight
            } else {
                __builtin_amdgcn_s_wait_tensorcnt(0);   // drain on last iter
            }
        }
#else
        // manual fallback: global -> LDS with the same pitched layouts
#pragma unroll
        for (int t = 0; t < 2; t++) {
            int idx = tid + t * 256;
            int row = idx >> 2;
            int c4  = (idx & 3) << 2;
            const float4 v = *(const float4*)(A + (size_t)(m0 + row) * 512 + k0 + c4);
            float* d = &As[buf][row * A_PITCH + c4];
            d[0] = v.x; d[1] = v.y; d[2] = v.z; d[3] = v.w;
        }
#pragma unroll
        for (int t = 0; t < 2; t++) {
            int idx = tid + t * 256;
            int row = idx >> 5;
            int c4  = (idx & 31) << 2;
            const float4 v = *(const float4*)(Bm + (size_t)(k0 + row) * 512 + n0 + c4);
            float* d = &Bs[buf][row * BP_ROW + c4];
            d[0] = v.x; d[1] = v.y; d[2] = v.z; d[3] = v.w;
        }
#endif
        __syncthreads();

        const float* as = &As[buf][0];
        const float* bs = &Bs[buf][0];
#pragma unroll
        for (int kk = 0; kk < 16; kk += 4) {
            v2f af[4], bf[2];
#pragma unroll
            for (int i = 0; i < 4; i++) {
                int ai = (waveM * 64 + i * 16 + r) * A_PITCH + kk + 2 * half;
                af[i].x = as[ai];
                af[i].y = as[ai + 1];
            }
#pragma unroll
            for (int j = 0; j < 2; j++) {
                int bi = (kk + 2 * half) * BP_ROW + waveN * 32 + j * 16 + r;
                bf[j].x = bs[bi];
                bf[j].y = bs[bi + BP_ROW];
            }
#pragma unroll
            for (int i = 0; i < 4; i++)
#pragma unroll
                for (int j = 0; j < 2; j++)
                    acc[i][j] = __builtin_amdgcn_wmma_f32_16x16x4_f32(
                        false, af[i], false, bf[j], (short)0, acc[i][j], false, false);
        }
        __syncthreads();
    }

#pragma unroll
    for (int i = 0; i < 4; i++)
#pragma unroll
        for (int j = 0; j < 2; j++)
#pragma unroll
            for (int v = 0; v < 8; v++) {
                int row = m0 + waveM * 64 + i * 16 + half * 8 + v;
                int col = n0 + waveN * 32 + j * 16 + r;
                Y[(size_t)row * 512 + col] = acc[i][j][v];
            }
}

// ---------------------------------------------------------------------------
// Col pass + fused PSD: Fr = Wr*Yr - Wi*Yi ; Fi = Wi*Yr + Wr*Yi
// Block tile 128x64, BK=16, 8 waves (4x2), wave tile 32x32, 4 accumulator sets
// ---------------------------------------------------------------------------
__global__ __launch_bounds__(256) void hf_colpass(const float* __restrict__ Yr,
                                                  const float* __restrict__ Yi,
                                                  const float* __restrict__ Wr,
                                                  const float* __restrict__ Wi,
                                                  float* __restrict__ partials) {
    __shared__ float Ar[2][128 * A_PITCH];
    __shared__ float Ai[2][128 * A_PITCH];
    __shared__ float Br[2][16 * BP_COL];
    __shared__ float Bi[2][16 * BP_COL];
    __shared__ float red[256];

    const int tid = threadIdx.x;
    const int img = blockIdx.z;
    const float* BYr = Yr + (size_t)img * IMG_PIX;
    const float* BYi = Yi + (size_t)img * IMG_PIX;

    const int m0 = blockIdx.y * 128;   // output row (freq j)
    const int n0 = blockIdx.x * 64;    // output col (freq k)

    const int lane  = tid & 31;
    const int wave  = tid >> 5;
    const int waveM = wave >> 1;   // 0..3
    const int waveN = wave & 1;    // 0..1
    const int half  = lane >> 4;
    const int r     = lane & 15;

    v8f arr[2][2], aii[2][2], ari[2][2], air[2][2];
#pragma unroll
    for (int i = 0; i < 2; i++)
#pragma unroll
        for (int j = 0; j < 2; j++) {
            arr[i][j] = hf_zero8(); aii[i][j] = hf_zero8();
            ari[i][j] = hf_zero8(); air[i][j] = hf_zero8();
        }

#if HF_HAS_TDM
    const bool issuer = (wave == 0);
    if (issuer) {   // prime buffer 0
        hf_tdm_load2d(Wr  + (size_t)m0 * 512, hf_lds_off(&Ar[0][0]),
                      16, 128, 512, 512, 512, 3, 0);
        hf_tdm_load2d(Wi  + (size_t)m0 * 512, hf_lds_off(&Ai[0][0]),
                      16, 128, 512, 512, 512, 3, 0);
        hf_tdm_load2d(BYr + n0,               hf_lds_off(&Br[0][0]),
                      64, 16, 512, 512, 512, 5, 7);
        hf_tdm_load2d(BYi + n0,               hf_lds_off(&Bi[0][0]),
                      64, 16, 512, 512, 512, 5, 7);
    }
#endif

    for (int k0 = 0; k0 < 512; k0 += 16) {
        const int buf = (k0 >> 4) & 1;
#if HF_HAS_TDM
        if (issuer) {
            if (k0 + 16 < 512) {
                hf_tdm_load2d(Wr  + (size_t)m0 * 512 + (k0 + 16),
                              hf_lds_off(&Ar[buf ^ 1][0]), 16, 128, 512, 512, 512, 3, 0);
                hf_tdm_load2d(Wi  + (size_t)m0 * 512 + (k0 + 16),
                              hf_lds_off(&Ai[buf ^ 1][0]), 16, 128, 512, 512, 512, 3, 0);
                hf_tdm_load2d(BYr + (size_t)(k0 + 16) * 512 + n0,
                              hf_lds_off(&Br[buf ^ 1][0]), 64, 16, 512, 512, 512, 5, 7);
                hf_tdm_load2d(BYi + (size_t)(k0 + 16) * 512 + n0,
                              hf_lds_off(&Bi[buf ^ 1][0]), 64, 16, 512, 512, 512, 5, 7);
                __builtin_amdgcn_s_wait_tensorcnt(4);   // next quad in flight
            } else {
                __builtin_amdgcn_s_wait_tensorcnt(0);   // drain on last iter
            }
        }
#else
#pragma unroll
        for (int t = 0; t < 2; t++) {
            int idx = tid + t * 256;
            int row = idx >> 2;
            int c4  = (idx & 3) << 2;
            size_t off = (size_t)(m0 + row) * 512 + k0 + c4;
            float4 vr = *(const float4*)(Wr + off);
            float4 vi = *(const float4*)(Wi + off);
            float* dr = &Ar[buf][row * A_PITCH + c4];
            float* di = &Ai[buf][row * A_PITCH + c4];
            dr[0]=vr.x; dr[1]=vr.y; dr[2]=vr.z; dr[3]=vr.w;
            di[0]=vi.x; di[1]=vi.y; di[2]=vi.z; di[3]=vi.w;
        }
        {
            int idx = tid;
            int row = idx >> 4;
            int c4  = (idx & 15) << 2;
            size_t off = (size_t)(k0 + row) * 512 + n0 + c4;
            float4 vr = *(const float4*)(BYr + off);
            float4 vi = *(const float4*)(BYi + off);
            float* dr = &Br[buf][row * BP_COL + c4];
            float* di = &Bi[buf][row * BP_COL + c4];
            dr[0]=vr.x; dr[1]=vr.y; dr[2]=vr.z; dr[3]=vr.w;
            di[0]=vi.x; di[1]=vi.y; di[2]=vi.z; di[3]=vi.w;
        }
#endif
        __syncthreads();

        const float* ar = &Ar[buf][0];
        const float* ai = &Ai[buf][0];
        const float* br = &Br[buf][0];
        const float* bi = &Bi[buf][0];
#pragma unroll
        for (int kk = 0; kk < 16; kk += 4) {
            v2f afr[2], afi[2], bfr[2], bfi[2];
#pragma unroll
            for (int i = 0; i < 2; i++) {
                int aidx = (waveM * 32 + i * 16 + r) * A_PITCH + kk + 2 * half;
                afr[i].x = ar[aidx]; afr[i].y = ar[aidx + 1];
                afi[i].x = ai[aidx]; afi[i].y = ai[aidx + 1];
            }
#pragma unroll
            for (int j = 0; j < 2; j++) {
                int bidx = (kk + 2 * half) * BP_COL + waveN * 32 + j * 16 + r;
                bfr[j].x = br[bidx]; bfr[j].y = br[bidx + BP_COL];
                bfi[j].x = bi[bidx]; bfi[j].y = bi[bidx + BP_COL];
            }
#pragma unroll
            for (int i = 0; i < 2; i++)
#pragma unroll
                for (int j = 0; j < 2; j++) {
                    arr[i][j] = __builtin_amdgcn_wmma_f32_16x16x4_f32(
                        false, afr[i], false, bfr[j], (short)0, arr[i][j], false, false);
                    aii[i][j] = __builtin_amdgcn_wmma_f32_16x16x4_f32(
                        false, afi[i], false, bfi[j], (short)0, aii[i][j], false, false);
                    ari[i][j] = __builtin_amdgcn_wmma_f32_16x16x4_f32(
                        false, afr[i], false, bfi[j], (short)0, ari[i][j], false, false);
                    air[i][j] = __builtin_amdgcn_wmma_f32_16x16x4_f32(
                        false, afi[i], false, bfr[j], (short)0, air[i][j], false, false);
                }
        }
        __syncthreads();
    }

    // Fused PSD epilogue: masked log10(|F|^2 + 1e-10)
    float lsum = 0.0f;
#pragma unroll
    for (int i = 0; i < 2; i++)
#pragma unroll
        for (int j = 0; j < 2; j++)
#pragma unroll
            for (int v = 0; v < 8; v++) {
                int jrow = m0 + waveM * 32 + i * 16 + half * 8 + v;   // freq j
                int kcol = n0 + waveN * 32 + j * 16 + r;              // freq k
                float fr = arr[i][j][v] - aii[i][j][v];
                float fi = air[i][j][v] + ari[i][j][v];
                bool lowJ = (jrow < 64) | (jrow >= 448);
                bool lowK = (kcol < 64) | (kcol >= 448);
                if (!(lowJ && lowK))
                    lsum += log10f(fr * fr + fi * fi + 1e-10f);
            }

    red[tid] = lsum;
    __syncthreads();
#pragma unroll
    for (int s = 128; s > 0; s >>= 1) {
        if (tid < s) red[tid] += red[tid + s];
        __syncthreads();
    }
    if (tid == 0)
        partials[img * 32 + blockIdx.y * 8 + blockIdx.x] = red[0];
}

// ---------------------------------------------------------------------------
// Final: loss = mean_b | psd_gen[b] - psd_tgt[b] |, psd = masked_sum / count
// ---------------------------------------------------------------------------
__global__ __launch_bounds__(64) void hf_final(const float* __restrict__ partials,
                                               float* __restrict__ out) {
    __shared__ float red[64];
    int b = threadIdx.x;
    float sg = 0.0f, st = 0.0f;
    for (int t = 0; t < 32; t++) {
        sg += partials[b * 32 + t];
        st += partials[2048 + b * 32 + t];
    }
    const float inv = 1.0f / HF_CNT;
    red[b] = fabsf(sg * inv - st * inv);
    __syncthreads();
    for (int s = 32; s > 0; s >>= 1) {
        if (b < s) red[b] += red[b + s];
        __syncthreads();
    }
    if (b == 0) out[0] = red[0] * (1.0f / 64.0f);   // WEIGHT == 1.0
}

// ---------------------------------------------------------------------------
extern "C" void kernel_launch(void* const* d_in, const int* in_sizes, int n_in,
                              void* d_out, int out_size, void* d_ws, size_t ws_size,
                              hipStream_t stream) {
    (void)in_sizes; (void)n_in; (void)out_size; (void)ws_size;
    const float* gen = (const float*)d_in[0];
    const float* tgt = (const float*)d_in[1];

    float* ws = (float*)d_ws;
    float* Wr = ws;                                  // 262144
    float* Wi = Wr + 262144;                         // 262144
    float* G  = Wi + 262144;                         // 16 * 262144
    float* Yr = G  + (size_t)CHUNK * IMG_PIX;        // 16 * 262144
    float* Yi = Yr + (size_t)CHUNK * IMG_PIX;        // 16 * 262144
    float* partials = Yi + (size_t)CHUNK * IMG_PIX;  // 4096

    hf_initW<<<1024, 256, 0, stream>>>(Wr, Wi);

    for (int t = 0; t < 2; t++) {
        const float* src = t ? tgt : gen;
        for (int c = 0; c < 64 / CHUNK; c++) {
            int imgBase = c * CHUNK;
            hf_gray<<<CHUNK * IMG_PIX / 256, 256, 0, stream>>>(src, imgBase, G);
            hf_rowpass<<<dim3(4, 4, CHUNK * 2), 256, 0, stream>>>(G, Wr, Wi, Yr, Yi);
            hf_colpass<<<dim3(8, 4, CHUNK), 256, 0, stream>>>(
                Yr, Yi, Wr, Wi, partials + t * 2048 + imgBase * 32);
        }
    }

    hf_final<<<1, 64, 0, stream>>>(partials, (float*)d_out);
}